// Harr_iwav_cat_76484777607458
// MI455X (gfx1250) — compile-verified
//
#include <hip/hip_runtime.h>
#include <hip/hip_bf16.h>

// Inverse-Haar 2x2 upsample of 4 subbands + concat with y.
// Shapes (fixed): B=8, D=32, H=128, W=128 -> out (8, 160, 256, 256) f32.
//
// Roofline: ~0.067 GFLOP vs ~470 MB first-touch traffic -> pure HBM-bound.
// Used inputs total 134 MB < 192 MB L2, so across graph replays we want the
// inputs L2-resident: RT (default) loads, NT stores for the 335 MB output
// (written once, never read -> evict-first, don't displace inputs).
// WMMA is inapplicable: zero reduction anywhere in the op.

typedef __attribute__((ext_vector_type(4))) float f4;

// Single fused kernel, 32768 blocks x 256 threads (8 wave32 waves/block):
//   blocks [0, 16384):     upsample LL/LH/HL/HH  (4,194,304 threads)
//   blocks [16384, 32768): stream-copy y         (4,194,304 threads)
__global__ __launch_bounds__(256) void haar_iwt_cat_kernel(
    const float* __restrict__ xLL,   // (8,32,128,128)
    const float* __restrict__ x,     // (8,4,32,128,128)
    const float* __restrict__ y,     // (8,32,256,256)
    float* __restrict__ out)         // (8,160,256,256)
{
    unsigned blk = blockIdx.x;

    if (blk < 16384u) {
        // ---- upsample: thread t = [s(2)|b(3)|d(5)|h(7)|w4(5)] ----
        unsigned t  = blk * 256u + threadIdx.x;
        unsigned w4 = t & 31u;            // W/4 = 32
        unsigned r  = t >> 5;
        unsigned h  = r & 127u;  r >>= 7; // H = 128
        unsigned d  = r & 31u;   r >>= 5; // D = 32
        unsigned b  = r & 7u;             // B = 8
        unsigned s  = r >> 3;             // subband 0..3 (0=LL from x_LL)

        // Branchless source select: no exec-mask divergence, just cndmask.
        unsigned q         = (s == 0u) ? b   : (b * 4u + s);
        const float* base  = (s == 0u) ? xLL : x;
        const float* src   = base
            + ((((size_t)q * 32u + d) * 128u + h) * 128u + (size_t)w4 * 4u);

        // gfx1250 stream-ahead prefetch (global_prefetch_b8); speculative.
        __builtin_prefetch(src + 4096, 0, 0);

        f4 v = *(const f4*)src;           // global_load_b128 (RT -> L2-resident)

        // 2x2 inverse-Haar taps (+-0.5):
        //   LL: ++/++   LH: -+/-+   HL: --/++   HH: +-/-+
        float a0 = (s == 1u || s == 2u) ? -0.5f : 0.5f;
        float b0 = (s >= 2u)            ? -0.5f : 0.5f;
        float a1 = (s == 1u || s == 3u) ? -0.5f : 0.5f;
        const float b1 = 0.5f;

        // Output: channel c = s*32 + d, spatial (2h, 8*w4), H2=W2=256.
        size_t obase = (((size_t)b * 160u + (size_t)s * 32u + d) * 256u
                        + (size_t)(h << 1)) * 256u + (size_t)w4 * 8u;
        float* r0 = out + obase;
        float* r1 = r0 + 256;

        f4 o;
        o.x = v.x * a0; o.y = v.x * b0; o.z = v.y * a0; o.w = v.y * b0;
        __builtin_nontemporal_store(o, (f4*)r0);         // global_store_b128 th:NT
        o.x = v.z * a0; o.y = v.z * b0; o.z = v.w * a0; o.w = v.w * b0;
        __builtin_nontemporal_store(o, (f4*)(r0 + 4));
        o.x = v.x * a1; o.y = v.x * b1; o.z = v.y * a1; o.w = v.y * b1;
        __builtin_nontemporal_store(o, (f4*)r1);
        o.x = v.z * a1; o.y = v.z * b1; o.z = v.w * a1; o.w = v.w * b1;
        __builtin_nontemporal_store(o, (f4*)(r1 + 4));
    } else {
        // ---- copy y into channels [128,160) of each batch ----
        unsigned t = (blk - 16384u) * 256u + threadIdx.x;
        unsigned b = t >> 19;             // 524,288 f4 per batch
        unsigned i = t & 0x7FFFFu;

        const f4* src = (const f4*)(y + (size_t)b * 2097152u) + i;     // 32*256*256
        f4*       dst = (f4*)(out + (size_t)b * 10485760u + 8388608u)  // b*160*65536 + 128*65536
                        + i;

        __builtin_prefetch((const float*)src + 4096, 0, 0);
        f4 v = *src;                                      // RT load: keep y in L2
        __builtin_nontemporal_store(v, dst);              // NT store
    }
}

extern "C" void kernel_launch(void* const* d_in, const int* in_sizes, int n_in,
                              void* d_out, int out_size, void* d_ws, size_t ws_size,
                              hipStream_t stream) {
    (void)in_sizes; (void)n_in; (void)out_size; (void)d_ws; (void)ws_size;

    const float* x_LL = (const float*)d_in[0];  // (8,32,128,128)
    const float* x    = (const float*)d_in[1];  // (8,4,32,128,128)
    const float* y    = (const float*)d_in[2];  // (8,32,256,256)
    float* out        = (float*)d_out;          // (8,160,256,256)

    haar_iwt_cat_kernel<<<32768, 256, 0, stream>>>(x_LL, x, y, out);
}